// GASF_81260781240507
// MI455X (gfx1250) — compile-verified
//
#include <hip/hip_runtime.h>
#include <math.h>

typedef __attribute__((ext_vector_type(2))) float v2f;
typedef __attribute__((ext_vector_type(8))) float v8f;

#define SEQ  128
#define FEAT 8
#define GEPS 1e-8f
#define LSTR 129   // padded LDS row stride (129 mod 64 == 1 -> conflict-free)

// One block per (b, imf). 256 threads = 8 waves; wave w owns s1-tile w.
// gasf[s1,s2,f] = c[f][s1]*c[f][s2] - s[f][s1]*s[f][s2], done as a K=16
// block-diagonal matmul via 4 chained V_WMMA_F32_16X16X4_F32.
__global__ __launch_bounds__(256) void gasf_wmma_kernel(const float* __restrict__ x,
                                                        float* __restrict__ out) {
  __shared__ float xs[SEQ * FEAT];     // raw x tile, [s][f]
  __shared__ float cc[FEAT * LSTR];    // cos(phi) = clipped scaled value, [f][s]
  __shared__ float sn[FEAT * LSTR];    // sin(phi) = sqrt(1 - c^2),        [f][s]
  __shared__ float scl[FEAT];          // 1/(2*std) or 1.0 if std == 0

  const int tid = threadIdx.x;
  const int bi  = blockIdx.x;                            // b*8 + imf
  const size_t in_base  = (size_t)bi * (SEQ * FEAT);
  const size_t out_base = (size_t)bi * ((size_t)SEQ * SEQ * FEAT);

  // ---- 1. cooperative load of x[b,i,:,:] (1024 floats) ----
  ((float4*)xs)[tid] = ((const float4*)(x + in_base))[tid];
  __syncthreads();

  // ---- 2. per-feature std over seq (two-pass, Bessel ddof=1) ----
  if (tid < FEAT) {
    float sum = 0.f;
    for (int s = 0; s < SEQ; ++s) sum += xs[s * FEAT + tid];
    const float mean = sum * (1.0f / SEQ);
    float var = 0.f;
    for (int s = 0; s < SEQ; ++s) {
      const float d = xs[s * FEAT + tid] - mean;
      var += d * d;
    }
    const float sd = sqrtf(var * (1.0f / (SEQ - 1)));
    scl[tid] = (sd > 0.f) ? (0.5f / sd) : 1.0f;
  }
  __syncthreads();

  // ---- 3. build c = clip(x*scale) and s = sqrt(1 - c^2) tables ----
  for (int e = tid; e < SEQ * FEAT; e += 256) {
    const int s = e / FEAT, f = e % FEAT;
    float v = xs[e] * scl[f];
    v = fminf(fmaxf(v, -1.0f + GEPS), 1.0f - GEPS);
    cc[f * LSTR + s] = v;
    sn[f * LSTR + s] = sqrtf(fmaxf(1.0f - v * v, 0.0f));
  }
  __syncthreads();

  // ---- 4. WMMA outer products ----
  const int wave = tid >> 5;          // s1 tile: rows [16w, 16w+16)
  const int lane = tid & 31;
  const int m    = lane & 15;         // M row (A) / N column (B,D)
  const int h    = lane >> 4;         // half-wave
  const int s1   = wave * 16 + m;

  // A fragments: per K-group g, lane holds A[m][4g+2h], A[m][4g+2h+1]
  //   k = f'*2 + t, f' = 2g+h  ->  { c[f'][s1], s[f'][s1] }
  v2f A[4];
#pragma unroll
  for (int g = 0; g < 4; ++g) {
    const int fp = 2 * g + h;
    A[g].x = cc[fp * LSTR + s1];
    A[g].y = sn[fp * LSTR + s1];
  }

  // B column for this lane: n = m = s2_off*8 + f
  const int f      = m & 7;
  const int s2off  = m >> 3;
  const int gsel   = f >> 1;                 // only group with nonzero B here
  const bool fpar  = ((f & 1) == h);         // and only if parity matches half

  float* const po = out + out_base + (size_t)(wave * 16 + 8 * h) * (SEQ * FEAT) + m;

  for (int s2p = 0; s2p < SEQ / 2; ++s2p) {  // 64 column tiles (s2 pairs)
    const int s2 = 2 * s2p + s2off;
    const float cv = cc[f * LSTR + s2];
    const float sv = sn[f * LSTR + s2];

    v8f acc = {};
#pragma unroll
    for (int g = 0; g < 4; ++g) {
      const bool sel = fpar && (g == gsel);
      v2f Bf;
      Bf.x = sel ?  cv : 0.0f;               // B[4g+2h  ][n] : t=0 -> +c
      Bf.y = sel ? -sv : 0.0f;               // B[4g+2h+1][n] : t=1 -> -s
      acc = __builtin_amdgcn_wmma_f32_16x16x4_f32(
          /*neg_a=*/false, A[g], /*neg_b=*/false, Bf,
          /*c_mod=*/(short)0, acc, /*reuse_a=*/false, /*reuse_b=*/false);
    }

    // D: lane holds D[v + 8h][m]; column offset within row = s2p*16 + m
    float* const pt = po + s2p * (2 * FEAT);
#pragma unroll
    for (int v = 0; v < 8; ++v) {
      __builtin_nontemporal_store(acc[v], pt + (size_t)v * (SEQ * FEAT));
    }
  }
}

extern "C" void kernel_launch(void* const* d_in, const int* in_sizes, int n_in,
                              void* d_out, int out_size, void* d_ws, size_t ws_size,
                              hipStream_t stream) {
  (void)in_sizes; (void)n_in; (void)d_ws; (void)ws_size; (void)out_size;
  const float* xp = (const float*)d_in[0];
  float* outp = (float*)d_out;
  // 64 batches * 8 imfs = 512 blocks, 256 threads each
  gasf_wmma_kernel<<<512, 256, 0, stream>>>(xp, outp);
}